// PointNetFPModule_12945031430506
// MI455X (gfx1250) — compile-verified
//
#include <hip/hip_runtime.h>
#include <hip/hip_bf16.h>

typedef __attribute__((ext_vector_type(2))) float v2f;
typedef __attribute__((ext_vector_type(8))) float v8f;

#define B_SZ 2
#define N_SZ 16384
#define M_SZ 4096
#define FCH 256
#define TCH 128
#define PCH 64

// ---------------------------------------------------------------------------
// Kernel 1: precompute per-ref float4 {-2x, -2y, -2z, |r|^2}
// ---------------------------------------------------------------------------
__global__ __launch_bounds__(256) void prep_ref_kernel(
    const float* __restrict__ ref_coords, float4* __restrict__ ref4) {
  int t = blockIdx.x * 256 + threadIdx.x;
  if (t >= B_SZ * M_SZ) return;
  float x = ref_coords[t * 3 + 0];
  float y = ref_coords[t * 3 + 1];
  float z = ref_coords[t * 3 + 2];
  float4 v;
  v.x = -2.0f * x;
  v.y = -2.0f * y;
  v.z = -2.0f * z;
  v.w = x * x + y * y + z * z;
  ref4[t] = v;
}

// sorted-3 smallest insertion: 5 min/max ops
__device__ __forceinline__ void ins3(int key, int& k0, int& k1, int& k2) {
  int h0 = max(key, k0); k0 = min(key, k0);
  int h1 = max(h0, k1);  k1 = min(h0, k1);
  k2 = min(h1, k2);
}

// ---------------------------------------------------------------------------
// Kernel 2: 3-NN search via V_WMMA_F32_16X16X4_F32.
// One wave handles 16 query points (columns). Loop over 256 tiles of 16 refs
// (rows). D[m][n] = (-2 r.q) + |r|^2 + |q|^2 = squared distance.
// Lane L (n = L&15, half = L>>4) holds rows {j + 8*half} in d[j].
// ---------------------------------------------------------------------------
__global__ __launch_bounds__(256) void nn3_kernel(
    const float* __restrict__ coords, const float4* __restrict__ ref4,
    int* __restrict__ idx3, float* __restrict__ w3) {
  const int lane = threadIdx.x & 31;
  const int wave = threadIdx.x >> 5;
  const int qt   = blockIdx.x * 8 + wave;   // global 16-query tile id
  const int tilesPerB = N_SZ >> 4;
  const int b = qt / tilesPerB;
  const int n = (qt - b * tilesPerB) * 16 + (lane & 15);
  const int half    = lane >> 4;            // 0 or 1
  const int row_off = half << 3;            // 0 or 8

  // query coords (both lane halves load the same query)
  const float* qc = coords + ((size_t)b * N_SZ + n) * 3;
  float qx = qc[0], qy = qc[1], qz = qc[2];
  float qn2 = qx * qx + qy * qy + qz * qz;

  // B matrix 4x16 (K x N): v0 = K(0 or 2), v1 = K(1 or 3) per lane half
  v2f bm;
  bm.x = half ? qz : qx;
  bm.y = half ? 1.0f : qy;

  // C matrix: |q_n|^2 broadcast across all rows
  v8f cm;
#pragma unroll
  for (int j = 0; j < 8; ++j) cm[j] = qn2;

  int k0 = 0x7FFFFFFF, k1 = 0x7FFFFFFF, k2 = 0x7FFFFFFF;

  const float4* rbase = ref4 + (size_t)b * M_SZ + (lane & 15);
#pragma unroll 2
  for (int t = 0; t < M_SZ; t += 16) {
    float4 r = rbase[t];  // ref index = t + (lane&15)
    // A matrix 16x4 (M x K): v0 = K(0 or 2), v1 = K(1 or 3) per lane half
    v2f am;
    am.x = half ? r.z : r.x;
    am.y = half ? r.w : r.y;
    v8f d = __builtin_amdgcn_wmma_f32_16x16x4_f32(
        false, am, false, bm, (short)0, cm, false, false);
    const int mo = t + row_off;  // tile base (mult of 16) | row_off(bit3) | j(0..2)
#pragma unroll
    for (int j = 0; j < 8; ++j) {
      int key = (__float_as_int(d[j]) & 0xFFFFF000) | (mo + j);
      ins3(key, k0, k1, k2);
    }
  }

  // merge the two lane-half top-3 lists via ds_bpermute (lane ^ 16)
  const int pidx = ((lane ^ 16) << 2);
  int p0 = __builtin_amdgcn_ds_bpermute(pidx, k0);
  int p1 = __builtin_amdgcn_ds_bpermute(pidx, k1);
  int p2 = __builtin_amdgcn_ds_bpermute(pidx, k2);
  ins3(p0, k0, k1, k2);
  ins3(p1, k0, k1, k2);
  ins3(p2, k0, k1, k2);

  if (half == 0) {
    int m0 = k0 & 0xFFF, m1 = k1 & 0xFFF, m2 = k2 & 0xFFF;
    // recompute exact squared distances from precomputed ref float4
    const float4* rb = ref4 + (size_t)b * M_SZ;
    float4 r0 = rb[m0], r1 = rb[m1], r2 = rb[m2];
    float d0 = fmaxf(qn2 + r0.x * qx + r0.y * qy + r0.z * qz + r0.w, 0.0f);
    float d1 = fmaxf(qn2 + r1.x * qx + r1.y * qy + r1.z * qz + r1.w, 0.0f);
    float d2 = fmaxf(qn2 + r2.x * qx + r2.y * qy + r2.z * qz + r2.w, 0.0f);
    float u0 = 1.0f / (d0 + 1e-8f);
    float u1 = 1.0f / (d1 + 1e-8f);
    float u2 = 1.0f / (d2 + 1e-8f);
    float inv = 1.0f / (u0 + u1 + u2);
    size_t o = ((size_t)b * N_SZ + n) * 3;
    idx3[o + 0] = m0; idx3[o + 1] = m1; idx3[o + 2] = m2;
    w3[o + 0] = u0 * inv; w3[o + 1] = u1 * inv; w3[o + 2] = u2 * inv;
  }
}

// ---------------------------------------------------------------------------
// Kernel 3: gather-interpolate + skip concat.
// grid.y encodes (b, chunk). 14 chunks of 32 channels:
//   0..7  -> features interp (256 ch), 8..11 -> t_embed interp (128 ch),
//   12..13 -> points_features copy (64 ch) into features channels 256..319.
// ---------------------------------------------------------------------------
__global__ __launch_bounds__(256) void interp_kernel(
    const float* __restrict__ refF, const float* __restrict__ refT,
    const float* __restrict__ ptsF, const int* __restrict__ idx3,
    const float* __restrict__ w3, float* __restrict__ out) {
  const int n  = blockIdx.x * 256 + threadIdx.x;
  const int bz = blockIdx.y;
  const int b  = bz / 14;
  const int cz = bz - b * 14;

  float* outF = out;                                   // [B,320,N]
  float* outT = out + (size_t)B_SZ * (FCH + PCH) * N_SZ;  // [B,128,N]

  if (cz >= 12) {
    int c0 = (cz - 12) * 32;
#pragma unroll 4
    for (int c = c0; c < c0 + 32; ++c)
      outF[((size_t)b * (FCH + PCH) + FCH + c) * N_SZ + n] =
          ptsF[((size_t)b * PCH + c) * N_SZ + n];
    return;
  }

  size_t q = ((size_t)b * N_SZ + n) * 3;
  int i0 = idx3[q + 0], i1 = idx3[q + 1], i2 = idx3[q + 2];
  float w0 = w3[q + 0], w1 = w3[q + 1], w2 = w3[q + 2];

  if (cz < 8) {
    int c0 = cz * 32;
    const float* base = refF + (size_t)b * FCH * M_SZ;
#pragma unroll 4
    for (int c = c0; c < c0 + 32; ++c) {
      const float* row = base + (size_t)c * M_SZ;
      outF[((size_t)b * (FCH + PCH) + c) * N_SZ + n] =
          w0 * row[i0] + w1 * row[i1] + w2 * row[i2];
    }
  } else {
    int c0 = (cz - 8) * 32;
    const float* base = refT + (size_t)b * TCH * M_SZ;
#pragma unroll 4
    for (int c = c0; c < c0 + 32; ++c) {
      const float* row = base + (size_t)c * M_SZ;
      outT[((size_t)b * TCH + c) * N_SZ + n] =
          w0 * row[i0] + w1 * row[i1] + w2 * row[i2];
    }
  }
}

extern "C" void kernel_launch(void* const* d_in, const int* in_sizes, int n_in,
                              void* d_out, int out_size, void* d_ws, size_t ws_size,
                              hipStream_t stream) {
  const float* coords     = (const float*)d_in[0];  // [B,N,3]
  const float* ref_coords = (const float*)d_in[1];  // [B,M,3]
  const float* ref_feats  = (const float*)d_in[2];  // [B,256,M]
  const float* ref_tembed = (const float*)d_in[3];  // [B,128,M]
  const float* pts_feats  = (const float*)d_in[4];  // [B,64,N]
  float* out = (float*)d_out;

  // workspace carve-up (256B-aligned regions)
  char* ws = (char*)d_ws;
  float4* ref4 = (float4*)ws;                         // B*M*16   = 128 KB
  size_t off = (size_t)B_SZ * M_SZ * sizeof(float4);
  off = (off + 255) & ~(size_t)255;
  int* idx3 = (int*)(ws + off);                       // B*N*3*4  = 384 KB
  off += (size_t)B_SZ * N_SZ * 3 * sizeof(int);
  off = (off + 255) & ~(size_t)255;
  float* w3 = (float*)(ws + off);                     // B*N*3*4  = 384 KB

  // 1) ref float4 precompute
  prep_ref_kernel<<<(B_SZ * M_SZ + 255) / 256, 256, 0, stream>>>(ref_coords, ref4);

  // 2) WMMA-based 3-NN search: one wave per 16 queries, 8 waves per block
  int nn_blocks = (B_SZ * N_SZ / 16) / 8;  // 2048 waves / 8 = 256 blocks
  nn3_kernel<<<nn_blocks, 256, 0, stream>>>(coords, ref4, idx3, w3);

  // 3) gather-interpolate + concat
  dim3 g(N_SZ / 256, B_SZ * 14);
  interp_kernel<<<g, 256, 0, stream>>>(ref_feats, ref_tembed, pts_feats,
                                       idx3, w3, out);
}